// CustomBBoxLoss_40630390620672
// MI455X (gfx1250) — compile-verified
//
#include <hip/hip_runtime.h>
#include <hip/hip_bf16.h>

// Problem constants (match reference)
#define HH      2048
#define WW      2048
#define NPIX    (HH * WW)      // 4,194,304 pixels per plane
#define NPLANES 6              // B*C = 2*3
#define NBOX    256

typedef __attribute__((ext_vector_type(2))) float v2f;
typedef __attribute__((ext_vector_type(4))) float v4f;
typedef __attribute__((ext_vector_type(8))) float v8f;

// ds_swizzle_b32 group-of-32 xor patterns: offset = xor_mask<<10 | and_mask(0x1f)
#define SWZ_XOR1  0x041F
#define SWZ_XOR2  0x081F
#define SWZ_XOR4  0x101F
#define SWZ_XOR8  0x201F

template <int PATTERN>
__device__ __forceinline__ float swz_add(float v) {
    int o = __builtin_amdgcn_ds_swizzle(__float_as_int(v), PATTERN);  // imm pattern
    return v + __int_as_float(o);
}

// ---------------------------------------------------------------------------
// Wave32 sum reduction via V_WMMA_F32_16X16X4_F32:
//   A = ones(16x4), B holds one f32 partial per lane (other B slot = 0).
//   D[m][n] = sum_k B[k][n]  ->  16 column sums, replicated in both wave
//   halves. A fixed 4-step ds_swizzle xor-butterfly (immediate patterns, no
//   index setup) finishes the 16-wide horizontal sum. Exact f32 accumulate.
// ---------------------------------------------------------------------------
__device__ __forceinline__ float wave_sum_wmma(float x) {
    v2f a; a.x = 1.0f; a.y = 1.0f;   // all-ones 16x4 A matrix
    v2f b; b.x = x;    b.y = 0.0f;   // one live B element per lane (32 of 64)
    v8f c = {};
    c = __builtin_amdgcn_wmma_f32_16x16x4_f32(
            /*neg_a=*/false, a, /*neg_b=*/false, b,
            /*c_mod=*/(short)0, c, /*reuse_a=*/false, /*reuse_b=*/false);
    float v = c[0];                  // column sums: lanes 0..15 (dup in 16..31)
    v = swz_add<SWZ_XOR1>(v);
    v = swz_add<SWZ_XOR2>(v);
    v = swz_add<SWZ_XOR4>(v);
    v = swz_add<SWZ_XOR8>(v);
    return v;                        // full 32-lane sum in every lane
}

// Block (256 threads = 8 waves) sum; result valid in all threads.
__device__ __forceinline__ float block_sum(float x, float* lds8) {
    float w = wave_sum_wmma(x);
    unsigned lane = threadIdx.x & 31u;
    unsigned wid  = threadIdx.x >> 5;
    __syncthreads();                 // protect lds reuse across back-to-back calls
    if (lane == 0) lds8[wid] = w;
    __syncthreads();
    float total = 0.0f;
#pragma unroll
    for (int i = 0; i < 8; ++i) total += lds8[i];
    return total;
}

// ---------------------------------------------------------------------------
// Pass 1: reduce over the 6 (B,C) planes -> per-pixel A = sum, A2 = sum of sq.
// pred streamed once with non-temporal loads (keep L2 free for A/A2, which
// pass 2 re-reads). 128-bit loads/stores, 4 pixels per thread.
// ---------------------------------------------------------------------------
__global__ void __launch_bounds__(256)
reduce_bc_kernel(const float* __restrict__ pred,
                 float* __restrict__ A, float* __restrict__ A2) {
    long base = ((long)blockIdx.x * blockDim.x + threadIdx.x) * 4;
    if (base >= NPIX) return;
    v4f s  = {0.0f, 0.0f, 0.0f, 0.0f};
    v4f s2 = {0.0f, 0.0f, 0.0f, 0.0f};
#pragma unroll
    for (int p = 0; p < NPLANES; ++p) {
        v4f v = __builtin_nontemporal_load((const v4f*)(pred + (long)p * NPIX + base));
        s  += v;
        s2 += v * v;
    }
    *(v4f*)(A  + base) = s;
    *(v4f*)(A2 + base) = s2;
}

// ---------------------------------------------------------------------------
// Pass 2: one workgroup per box. Flattened-region striding (full utilization
// even for narrow boxes), then WMMA block reduction. A/A2 live in L2.
// ---------------------------------------------------------------------------
__global__ void __launch_bounds__(256)
box_loss_kernel(const float* __restrict__ A, const float* __restrict__ A2,
                const int* __restrict__ box_y, const int* __restrict__ box_x,
                const int* __restrict__ box_h, const int* __restrict__ box_w,
                const int* __restrict__ box_cls,
                float* __restrict__ perbox) {
    __shared__ float lds8[8];
    const int b  = blockIdx.x;
    const int y0 = box_y[b];
    const int x0 = box_x[b];
    const int h  = box_h[b];
    const int w  = box_w[b];
    const unsigned total = (unsigned)(h * w);

    float s1 = 0.0f, s2 = 0.0f;
    for (unsigned i = threadIdx.x; i < total; i += 256u) {
        unsigned r   = i / (unsigned)w;
        unsigned col = i - r * (unsigned)w;
        long idx = (long)(y0 + (int)r) * WW + (long)(x0 + (int)col);
        s1 += A[idx];
        s2 += A2[idx];
    }

    float S1 = block_sum(s1, lds8);
    float S2 = block_sum(s2, lds8);

    if (threadIdx.x == 0) {
        float cls = (float)box_cls[b];
        float cnt = (float)NPLANES * (float)(h * w);
        perbox[b] = (S2 - 2.0f * cls * S1 + cls * cls * cnt) / cnt;
    }
}

// ---------------------------------------------------------------------------
// Pass 3: mean over the 256 per-box losses (no atomics -> deterministic).
// ---------------------------------------------------------------------------
__global__ void __launch_bounds__(256)
final_mean_kernel(const float* __restrict__ perbox, float* __restrict__ out) {
    __shared__ float lds8[8];
    float x = perbox[threadIdx.x];          // exactly 256 threads, 256 boxes
    float S = block_sum(x, lds8);
    if (threadIdx.x == 0) out[0] = S * (1.0f / (float)NBOX);
}

// ---------------------------------------------------------------------------
extern "C" void kernel_launch(void* const* d_in, const int* in_sizes, int n_in,
                              void* d_out, int out_size, void* d_ws, size_t ws_size,
                              hipStream_t stream) {
    const float* pred    = (const float*)d_in[0];
    const int*   box_y   = (const int*)d_in[1];
    const int*   box_x   = (const int*)d_in[2];
    const int*   box_h   = (const int*)d_in[3];
    const int*   box_w   = (const int*)d_in[4];
    const int*   box_cls = (const int*)d_in[5];
    float*       out     = (float*)d_out;

    // Workspace layout: A [NPIX] | A2 [NPIX] | perbox [NBOX]   (~33.6 MB)
    float* A      = (float*)d_ws;
    float* A2     = A + NPIX;
    float* perbox = A2 + NPIX;

    reduce_bc_kernel<<<NPIX / 4 / 256, 256, 0, stream>>>(pred, A, A2);
    box_loss_kernel<<<NBOX, 256, 0, stream>>>(A, A2, box_y, box_x, box_h, box_w,
                                              box_cls, perbox);
    final_mean_kernel<<<1, 256, 0, stream>>>(perbox, out);
}